// MultiLabelRankingLoss_7997229105161
// MI455X (gfx1250) — compile-verified
//
#include <hip/hip_runtime.h>
#include <stdint.h>

// CDNA5 wave32 WMMA vector operand types
typedef __attribute__((ext_vector_type(2))) float v2f;  // f32 A/B: 16x4 / 4x16 = 2 VGPRs/lane
typedef __attribute__((ext_vector_type(8))) float v8f;  // f32 C/D: 16x16     = 8 VGPRs/lane

#define MARGIN   1.0f
#define C_DIM    1024
#define TILES    (C_DIM / 16)   // 64 tiles of 16 along the candidate axis
#define BLOCK    256            // 8 wave32 waves per workgroup
#define PAD      16             // zero pad read by lanes 16..31 (K=2,3 -> 0)

// One workgroup per batch row. Computes
//   partial_num[b] = sum_{n,p} relu( neg_n * pos_p * (s_n - s_p + MARGIN) )
//   partial_den[b] = n_pos * (C - n_pos)
// The 16x16 outer-difference tile is one V_WMMA_F32_16X16X4_F32 rank-2 update:
//   D = A(16x4) x B(4x16),  A=[s*neg, neg, 0, 0],  B=[pos; (m-s)*pos; 0; 0].
// Masks are baked into A/B (relu(m*x) = m*relu(x) for m in {0,1}), so the only
// post-WMMA work is relu + accumulate. Lanes 16..31 (which carry K=2,3 of the
// operands) are steered to a zeroed LDS pad, so no masking VALU in the loop.
__global__ __launch_bounds__(BLOCK) void rankloss_rows_kernel(
    const float* __restrict__ scores,      // [B, C] f32
    const uint8_t* __restrict__ pos_mask,  // [B, C] bool (1 byte)
    float* __restrict__ partial_num,       // [B]
    float* __restrict__ partial_den)       // [B]
{
    __shared__ v2f sA[C_DIM + PAD];  // {s*neg, neg}          -> A columns K=0,1
    __shared__ v2f sB[C_DIM + PAD];  // {pos, (m-s)*pos}      -> B rows    K=0,1
    __shared__ float red[BLOCK];

    const int b = blockIdx.x;
    const int t = threadIdx.x;
    const float*   row = scores   + (size_t)b * C_DIM;
    const uint8_t* pm  = pos_mask + (size_t)b * C_DIM;

    // gfx1250 prefetch path (global_prefetch_b8)
    __builtin_prefetch(row + t, 0, 0);

    // ---- stage masked operand planes into LDS, count positives ----
    float npos_local = 0.0f;
    for (int c = t; c < C_DIM; c += BLOCK) {
        float s = row[c];
        float p = pm[c] ? 1.0f : 0.0f;
        float n = 1.0f - p;
        sA[c] = (v2f){s * n, n};
        sB[c] = (v2f){p, (MARGIN - s) * p};
        npos_local += p;
    }
    if (t < PAD) {                       // zero pad: source for lanes 16..31
        sA[C_DIM + t] = (v2f){0.0f, 0.0f};
        sB[C_DIM + t] = (v2f){0.0f, 0.0f};
    }
    red[t] = npos_local;
    __syncthreads();
    for (int stride = BLOCK / 2; stride > 0; stride >>= 1) {
        if (t < stride) red[t] += red[t + stride];
        __syncthreads();
    }
    const float n_pos = red[0];
    __syncthreads();  // red[] reused below

    // ---- WMMA tile sweep ----
    const int wave = t >> 5;             // 0..7
    const int lane = t & 31;
    const int idx  = lane & 15;
    const bool lo  = (lane < 16);        // lanes 0..15 carry K=0,1; 16..31 -> pad
    const int abase = lo ? idx : (C_DIM + idx);
    const int bbase = lo ? idx : (C_DIM + idx);
    const int astep = lo ? 16 : 0;       // per-lane stride: address does the masking
    const int bstep = lo ? 16 : 0;

    v8f accv = {};                       // 8 independent relu-accumulate chains
    for (int i = wave; i < TILES; i += (BLOCK / 32)) {
        const v2f a = sA[abase + i * astep];          // one ds_load_b64
        int boff = bbase;
        for (int j = 0; j < TILES; ++j, boff += bstep) {
            const v2f bb = sB[boff];                  // one ds_load_b64
            v8f cz = {};
            // (neg_a, A, neg_b, B, c_mod, C, reuse_a, reuse_b)
            v8f d = __builtin_amdgcn_wmma_f32_16x16x4_f32(
                false, a, false, bb, (short)0, cz, false, false);
#pragma unroll
            for (int r = 0; r < 8; ++r) {
                // relu(x) = (x + |x|) * 0.5 : |x| is a free src modifier on the
                // add, *0.5 folds into v_fmac -> 2 VALU per element, no
                // canonicalizing double-max.
                float x = d[r];
                accv[r] = __builtin_fmaf(x + __builtin_fabsf(x), 0.5f, accv[r]);
            }
        }
    }
    float acc = ((accv[0] + accv[1]) + (accv[2] + accv[3])) +
                ((accv[4] + accv[5]) + (accv[6] + accv[7]));

    // ---- deterministic block reduction of the numerator ----
    red[t] = acc;
    __syncthreads();
    for (int stride = BLOCK / 2; stride > 0; stride >>= 1) {
        if (t < stride) red[t] += red[t + stride];
        __syncthreads();
    }
    if (t == 0) {
        partial_num[b] = red[0];
        partial_den[b] = n_pos * ((float)C_DIM - n_pos);
    }
}

// Fixed-order final reduction: bitwise deterministic across replays.
__global__ void rankloss_final_kernel(const float* __restrict__ partial_num,
                                      const float* __restrict__ partial_den,
                                      float* __restrict__ out, int nrows)
{
    if (blockIdx.x == 0 && threadIdx.x == 0) {
        float num = 0.0f, den = 0.0f;
        for (int i = 0; i < nrows; ++i) {
            num += partial_num[i];
            den += partial_den[i];
        }
        out[0] = (den == 0.0f) ? 0.0f : (num / den);
    }
}

extern "C" void kernel_launch(void* const* d_in, const int* in_sizes, int n_in,
                              void* d_out, int out_size, void* d_ws, size_t ws_size,
                              hipStream_t stream)
{
    (void)n_in; (void)out_size; (void)ws_size;
    const float*   scores   = (const float*)d_in[0];
    const uint8_t* pos_mask = (const uint8_t*)d_in[1];  // jax bool = 1 byte/elem
    float* out = (float*)d_out;

    const int B = in_sizes[0] / C_DIM;   // 128 for the reference shapes

    float* partial_num = (float*)d_ws;
    float* partial_den = partial_num + B;

    rankloss_rows_kernel<<<B, BLOCK, 0, stream>>>(scores, pos_mask,
                                                  partial_num, partial_den);
    rankloss_final_kernel<<<1, 32, 0, stream>>>(partial_num, partial_den, out, B);
}